// CRF_69552700392038
// MI455X (gfx1250) — compile-verified
//
#include <hip/hip_runtime.h>
#include <math.h>

typedef __attribute__((ext_vector_type(2))) float v2f;
typedef __attribute__((ext_vector_type(8))) float v8f;

#define TLEN 512
#define LDIM 128
#define AN_STRIDE 132   // pad alphaNew rows: rows b and b+8 differ by 32 banks

__device__ __forceinline__ float halfwave_max(float v) {
    // reduce across the 16 lanes of a half-wave (one batch row)
    v = fmaxf(v, __shfl_xor(v, 1, 16));
    v = fmaxf(v, __shfl_xor(v, 2, 16));
    v = fmaxf(v, __shfl_xor(v, 4, 16));
    v = fmaxf(v, __shfl_xor(v, 8, 16));
    return v;
}
__device__ __forceinline__ float halfwave_sum(float v) {
    v += __shfl_xor(v, 1, 16);
    v += __shfl_xor(v, 2, 16);
    v += __shfl_xor(v, 4, 16);
    v += __shfl_xor(v, 8, 16);
    return v;
}

__global__ __launch_bounds__(256, 1)
void crf_fwd_kernel(const float* __restrict__ y_true,
                    const float* __restrict__ y_pred,
                    const float* __restrict__ trans,
                    float* __restrict__ out)
{
    __shared__ float alphaE[LDIM * 16];         // exp(alpha - shift), TRANSPOSED [k][b]
    __shared__ float alphaNew[16 * AN_STRIDE];  // [b][j] log(dot) tiles (padded)
    __shared__ float lognorm[16];
    __shared__ float red[256];                  // scratch for score reduction
    __shared__ int   lab[16 * TLEN];            // decoded one-hot labels

    const int tid    = threadIdx.x;
    const int lane   = tid & 31;
    const int wave   = tid >> 5;          // j-tile index 0..7
    const int bBase  = blockIdx.x * 16;   // 16 batch groups

    const int laneLo  = lane & 15;
    const int laneHiK = (lane >> 4) << 1; // K offset of upper lane-half (A/B frags)
    const int laneHiM = (lane >> 4) << 3; // M offset of upper lane-half (C/D frags)

    // ---- B operand: exp(trans) tile 128 x 16, resident in VGPRs for whole scan
    v2f bfrag[32];
    {
        const int col = wave * 16 + laneLo;
        #pragma unroll
        for (int kc = 0; kc < 32; ++kc) {
            const int k0 = kc * 4 + laneHiK;
            bfrag[kc].x = __expf(trans[(k0 + 0) * LDIM + col]);
            bfrag[kc].y = __expf(trans[(k0 + 1) * LDIM + col]);
        }
    }

    // reduction-phase roles: thread -> (batch row bRed, j-segment of 8)
    // the 16 threads of batch bRed are one half-wave -> shuffle reductions
    const int bRed = tid >> 4;
    const int seg  = tid & 15;
    const int j0   = seg * 8;
    const float* ypB = y_pred + (size_t)(bBase + bRed) * TLEN * LDIM;

    // per-thread (per-batch-row) shift state, fully register resident
    float shPrev, shCur;

    // ---- init: alpha(0) = y_pred[:,0,:]  (exact shift, no LDS round-trip)
    {
        float v[8];
        float lm = -INFINITY;
        #pragma unroll
        for (int q = 0; q < 8; ++q) {
            v[q] = ypB[j0 + q];
            lm = fmaxf(lm, v[q]);
        }
        const float m = halfwave_max(lm);
        shCur = m;          // shift baked into alphaE(0)
        shPrev = m;         // carry for step 1
        #pragma unroll
        for (int q = 0; q < 8; ++q)
            alphaE[(j0 + q) * 16 + bRed] = __expf(v[q] - m);
    }
    __syncthreads();

    // ---- forward scan: 511 dependent steps of 16x128 @ 128x16 fp32 WMMA
    for (int t = 1; t < TLEN; ++t) {
        // Hoist this step's emission into registers: loads overlap the WMMA chain.
        float ev[8];
        {
            const float* emit = ypB + (size_t)t * LDIM + j0;
            #pragma unroll
            for (int q = 0; q < 8; ++q) ev[q] = emit[q];
        }
        // Warm L2 for the post-scan label pass (4 lines per y_true row).
        if (tid < 64) {
            const float* yrow = y_true +
                ((size_t)(bBase + (tid >> 2)) * TLEN + (t - 1)) * LDIM + (tid & 3) * 32;
            __builtin_prefetch(yrow, 0, 1);
        }

        // Preload ALL A-fragments, then fence the scheduler so the whole DS
        // burst issues before the WMMA chain (pay LDS latency once, not 32x).
        v2f afr[32];
        #pragma unroll
        for (int kc = 0; kc < 32; ++kc) {
            const int ka = (kc * 4 + laneHiK) * 16 + laneLo;
            afr[kc].x = alphaE[ka];        // row k,   col b  (stride 16: conflict-free)
            afr[kc].y = alphaE[ka + 16];   // row k+1, col b
        }
        __builtin_amdgcn_sched_barrier(0);
        v8f acc = {};
        #pragma unroll
        for (int kc = 0; kc < 32; ++kc) {
            acc = __builtin_amdgcn_wmma_f32_16x16x4_f32(
                      false, afr[kc], false, bfrag[kc], (short)0, acc, false, false);
        }
        // D frag -> log-space tile
        #pragma unroll
        for (int r = 0; r < 8; ++r) {
            const int b = r + laneHiM;
            alphaNew[b * AN_STRIDE + wave * 16 + laneLo] = __logf(acc[r]);
        }
        __syncthreads();

        // Fused pass: alpha = log(dot) + carry + emit; alphaE = exp(alpha - shCur);
        // new shift via half-wave shuffle max (lagged shift: drift ~ log L + maxima).
        {
            float lm = -INFINITY;
            #pragma unroll
            for (int q = 0; q < 8; ++q) {
                const int j = j0 + q;
                const float v = alphaNew[bRed * AN_STRIDE + j] + shPrev + ev[q];
                alphaE[j * 16 + bRed] = __expf(v - shCur);
                lm = fmaxf(lm, v);
            }
            const float m = halfwave_max(lm);
            shPrev = shCur;   // shift baked into alphaE just built
            shCur  = m;       // lagged shift for the next step
        }
        __syncthreads();
    }

    // ---- log_norm[b] = shift_used + log(sum_j exp(alpha_j - shift_used))
    {
        float s = 0.f;
        #pragma unroll
        for (int q = 0; q < 8; ++q) s += alphaE[(j0 + q) * 16 + bRed];
        s = halfwave_sum(s);
        if (seg == 0) lognorm[bRed] = shPrev + __logf(s);
    }
    __syncthreads();

    // ---- decode one-hot labels (single full pass over y_true, L2-warm)
    for (int p = tid; p < 16 * TLEN; p += 256) {
        const int b  = p >> 9;          // /512
        const int tt = p & (TLEN - 1);
        const float4* row =
            (const float4*)(y_true + ((size_t)(bBase + b) * TLEN + tt) * LDIM);
        int l = 0;
        #pragma unroll 4
        for (int j4 = 0; j4 < LDIM / 4; ++j4) {
            const float4 v = row[j4];
            if (v.x > 0.5f) l = j4 * 4 + 0;
            if (v.y > 0.5f) l = j4 * 4 + 1;
            if (v.z > 0.5f) l = j4 * 4 + 2;
            if (v.w > 0.5f) l = j4 * 4 + 3;
        }
        lab[p] = l;
    }
    __syncthreads();

    // ---- point + transition scores (gather), then output
    {
        float acc_s = 0.f;
        const int t0 = seg * 32;
        #pragma unroll 4
        for (int q = 0; q < 32; ++q) {
            const int tt = t0 + q;
            const int l0 = lab[bRed * TLEN + tt];
            acc_s += ypB[(size_t)tt * LDIM + l0];
            if (tt < TLEN - 1) {
                const int l1 = lab[bRed * TLEN + tt + 1];
                acc_s += trans[l0 * LDIM + l1];
            }
        }
        red[tid] = acc_s;
    }
    __syncthreads();
    if (tid < 16) {
        float s = 0.f;
        #pragma unroll
        for (int i = 0; i < 16; ++i) s += red[tid * 16 + i];
        out[bBase + tid] = lognorm[tid] - s;
    }
}

extern "C" void kernel_launch(void* const* d_in, const int* in_sizes, int n_in,
                              void* d_out, int out_size, void* d_ws, size_t ws_size,
                              hipStream_t stream) {
    const float* y_true = (const float*)d_in[0];
    const float* y_pred = (const float*)d_in[1];
    const float* trans  = (const float*)d_in[2];
    float* out = (float*)d_out;
    (void)in_sizes; (void)n_in; (void)out_size; (void)d_ws; (void)ws_size;
    crf_fwd_kernel<<<16, 256, 0, stream>>>(y_true, y_pred, trans, out);
}